// SSDLoss_9844065042824
// MI455X (gfx1250) — compile-verified
//
#include <hip/hip_runtime.h>
#include <math.h>

// ---------------- problem constants ----------------
#define B_IMG 32
#define NGT   32
#define A_N   8732
#define C_N   81
#define THRESH 0.4f

#define BLK   256
#define NBLK_A ((A_N + BLK - 1) / BLK)   // 35

typedef __attribute__((ext_vector_type(2))) float v2f;
typedef __attribute__((ext_vector_type(8))) float v8f;

// ---------------- K1: per-anchor best gt (gt_overlap / gt_idx) ----------------
__global__ void k_iou_anchor(const float* __restrict__ tar_bbs,
                             const float* __restrict__ anchors,
                             float* __restrict__ gt_ovl,
                             int* __restrict__ gt_idx) {
    const int b = blockIdx.y;
    const int a = blockIdx.x * BLK + threadIdx.x;
    if (a >= A_N) return;

    const float4 bb = ((const float4*)anchors)[a];
    const float area_b = (bb.z - bb.x) * (bb.w - bb.y);
    const float4* tb = (const float4*)(tar_bbs + (size_t)b * NGT * 4);

    float best = -1.0f; int bi = 0;
    #pragma unroll 4
    for (int g = 0; g < NGT; ++g) {
        float4 ab = tb[g];
        float area_a = (ab.z - ab.x) * (ab.w - ab.y);
        float ltx = fmaxf(ab.x, bb.x), lty = fmaxf(ab.y, bb.y);
        float rbx = fminf(ab.z, bb.z), rby = fminf(ab.w, bb.w);
        float w = fmaxf(rbx - ltx, 0.0f), h = fmaxf(rby - lty, 0.0f);
        float inter = w * h;
        float iou = inter / (area_a + area_b - inter);
        if (iou > best) { best = iou; bi = g; }   // first index wins ties (jnp.argmax)
    }
    gt_ovl[(size_t)b * A_N + a] = best;
    gt_idx[(size_t)b * A_N + a] = bi;
}

// ---------------- K2: per-gt best anchor (prior_idx), one wave per (b,g) ----------------
__global__ void k_prior(const float* __restrict__ tar_bbs,
                        const float* __restrict__ anchors,
                        int* __restrict__ prior_idx) {
    const int g = blockIdx.x;
    const int b = blockIdx.y;
    const int lane = threadIdx.x;   // 32 lanes = 1 wave

    const float4 ab = ((const float4*)tar_bbs)[(size_t)b * NGT + g];
    const float area_a = (ab.z - ab.x) * (ab.w - ab.y);

    float best = -1.0f; int bi = 0x7FFFFFFF;
    for (int a = lane; a < A_N; a += 32) {
        float4 bb = ((const float4*)anchors)[a];
        float area_b = (bb.z - bb.x) * (bb.w - bb.y);
        float ltx = fmaxf(ab.x, bb.x), lty = fmaxf(ab.y, bb.y);
        float rbx = fminf(ab.z, bb.z), rby = fminf(ab.w, bb.w);
        float w = fmaxf(rbx - ltx, 0.0f), h = fmaxf(rby - lty, 0.0f);
        float inter = w * h;
        float iou = inter / (area_a + area_b - inter);
        if (iou > best) { best = iou; bi = a; }   // ascending a + strict '>' keeps smallest index
    }
    // lexicographic (max value, min index) wave reduction -> exact jnp.argmax tie rule
    #pragma unroll
    for (int off = 16; off >= 1; off >>= 1) {
        float ob = __shfl_xor(best, off, 32);
        int   oi = __shfl_xor(bi,   off, 32);
        if (ob > best || (ob == best && oi < bi)) { best = ob; bi = oi; }
    }
    if (lane == 0) prior_idx[b * NGT + g] = bi;
}

// ---------------- K3: forced-match overrides (serial per image => deterministic) ----------------
__global__ void k_override(const int* __restrict__ prior_idx,
                           float* __restrict__ gt_ovl,
                           int* __restrict__ gt_idx) {
    const int b = threadIdx.x;      // 32 threads, one per image
    if (b >= B_IMG) return;
    for (int g = 0; g < NGT; ++g) { // ascending g: later (larger) g overwrites == forced.max()
        int a = prior_idx[b * NGT + g];
        gt_ovl[(size_t)b * A_N + a] = 1.99f;
        gt_idx[(size_t)b * A_N + a] = g;
    }
}

// ---------------- K4: loc L1 + CE per anchor, deterministic block reduction ----------------
__global__ void k_loss(const float* __restrict__ pred_bbs,
                       const float* __restrict__ pred_cs,
                       const float* __restrict__ tar_bbs,
                       const int*   __restrict__ tar_c,
                       const float* __restrict__ anchors,
                       const float* __restrict__ grid_size,
                       const float* __restrict__ gt_ovl,
                       const int*   __restrict__ gt_idx,
                       float* __restrict__ p_loc,
                       float* __restrict__ p_nll,
                       int*   __restrict__ p_np) {
    const int b = blockIdx.y;
    const int a = blockIdx.x * BLK + threadIdx.x;

    float loc = 0.0f, nll = 0.0f; int np = 0;
    if (a < A_N) {
        const float ovl = gt_ovl[(size_t)b * A_N + a];
        const int   gi  = gt_idx[(size_t)b * A_N + a];
        const bool  pos = ovl > THRESH;
        int label = 0;
        if (pos) {
            np = 1;
            label = tar_c[b * NGT + gi];
            float4 pb = ((const float4*)pred_bbs)[(size_t)b * A_N + a];
            float4 an = ((const float4*)anchors)[a];
            float  gs = grid_size[a] * 0.5f;
            float4 m  = ((const float4*)tar_bbs)[(size_t)b * NGT + gi];
            float x0 = an.x + tanhf(pb.x) * gs;
            float y0 = an.y + tanhf(pb.y) * gs;
            float x1 = an.z + tanhf(pb.z) * gs;
            float y1 = an.w + tanhf(pb.w) * gs;
            loc = fabsf(x0 - m.x) + fabsf(y0 - m.y) + fabsf(x1 - m.z) + fabsf(y1 - m.w);
        }
        // log_softmax over 81 classes: two passes, second pass hits L0 (row = 324B)
        const float* row = pred_cs + ((size_t)b * A_N + a) * C_N;
        float mx = -3.0e38f;
        for (int c = 0; c < C_N; ++c) mx = fmaxf(mx, row[c]);
        float s = 0.0f;
        for (int c = 0; c < C_N; ++c) s += expf(row[c] - mx);
        nll = -(row[label] - mx - logf(s));
    }

    __shared__ float s_loc[BLK];
    __shared__ float s_nll[BLK];
    __shared__ int   s_np[BLK];
    const int t = threadIdx.x;
    s_loc[t] = loc; s_nll[t] = nll; s_np[t] = np;
    __syncthreads();
    for (int off = BLK / 2; off > 0; off >>= 1) {   // fixed tree => deterministic
        if (t < off) {
            s_loc[t] += s_loc[t + off];
            s_nll[t] += s_nll[t + off];
            s_np[t]  += s_np[t + off];
        }
        __syncthreads();
    }
    if (t == 0) {
        p_loc[b * NBLK_A + blockIdx.x] = s_loc[0];
        p_nll[b * NBLK_A + blockIdx.x] = s_nll[0];
        p_np [b * NBLK_A + blockIdx.x] = s_np[0];
    }
}

// ---------------- K5: finalize per-image loss; WMMA f32 16x16x4 wave reduction ----------------
__global__ void k_final(const float* __restrict__ p_loc,
                        const float* __restrict__ p_nll,
                        const int*   __restrict__ p_np,
                        float* __restrict__ out) {
    const int b = threadIdx.x;      // exactly 32 threads = 1 full wave (EXEC all 1s for WMMA)
    float loc = 0.0f, nll = 0.0f; int np = 0;
    for (int k = 0; k < NBLK_A; ++k) {
        loc += p_loc[b * NBLK_A + k];
        nll += p_nll[b * NBLK_A + k];
        np  += p_np [b * NBLK_A + k];
    }
    int denom = np * 4; if (denom < 1) denom = 1;
    float loss = loc / (float)denom + nll / (float)A_N;

    // Cross-lane sum of 32 per-image losses via V_WMMA_F32_16X16X4_F32.
    // A layout (16x4 f32): lanes 0-15 hold M=lane {K0=v0,K1=v1}; lanes 16-31 hold {K2=v0,K3=v1}.
    // With a.y = 0 and B = all-ones: D[m,n] = loss[m] + loss[m+16] for every n.
    v2f av; av.x = loss; av.y = 0.0f;
    v2f bv; bv.x = 1.0f; bv.y = 1.0f;
    v8f cv = {};
    v8f dv = __builtin_amdgcn_wmma_f32_16x16x4_f32(
        /*neg_a=*/false, av, /*neg_b=*/false, bv,
        /*c_mod=*/(short)0, cv, /*reuse_a=*/false, /*reuse_b=*/false);
    // D VGPR r: lanes 0-15 hold M=r, lanes 16-31 hold M=r+8. Sum the 8 VGPRs,
    // then fold lane-halves: every lane ends with sum over all 32 images.
    float v = dv[0] + dv[1] + dv[2] + dv[3] + dv[4] + dv[5] + dv[6] + dv[7];
    v += __shfl_xor(v, 16, 32);
    if (b == 0) out[0] = v;
}

// ---------------- launcher ----------------
extern "C" void kernel_launch(void* const* d_in, const int* in_sizes, int n_in,
                              void* d_out, int out_size, void* d_ws, size_t ws_size,
                              hipStream_t stream) {
    const float* pred_bbs  = (const float*)d_in[0];   // [32,8732,4]
    const float* pred_cs   = (const float*)d_in[1];   // [32,8732,81]
    const float* tar_bbs   = (const float*)d_in[2];   // [32,32,4]
    const int*   tar_c     = (const int*)  d_in[3];   // [32,32]
    const float* anchors   = (const float*)d_in[4];   // [8732,4]
    const float* grid_size = (const float*)d_in[5];   // [8732,1]
    float* out = (float*)d_out;

    // workspace layout (all 16B aligned)
    char* ws = (char*)d_ws;
    size_t off = 0;
    float* gt_ovl    = (float*)(ws + off); off += (size_t)B_IMG * A_N * sizeof(float);
    int*   gt_idx    = (int*)  (ws + off); off += (size_t)B_IMG * A_N * sizeof(int);
    int*   prior_idx = (int*)  (ws + off); off += (size_t)B_IMG * NGT * sizeof(int);
    float* p_loc     = (float*)(ws + off); off += (size_t)B_IMG * NBLK_A * sizeof(float);
    float* p_nll     = (float*)(ws + off); off += (size_t)B_IMG * NBLK_A * sizeof(float);
    int*   p_np      = (int*)  (ws + off); off += (size_t)B_IMG * NBLK_A * sizeof(int);
    (void)in_sizes; (void)n_in; (void)out_size; (void)ws_size;

    k_iou_anchor<<<dim3(NBLK_A, B_IMG), BLK, 0, stream>>>(tar_bbs, anchors, gt_ovl, gt_idx);
    k_prior     <<<dim3(NGT, B_IMG), 32, 0, stream>>>(tar_bbs, anchors, prior_idx);
    k_override  <<<1, B_IMG, 0, stream>>>(prior_idx, gt_ovl, gt_idx);
    k_loss      <<<dim3(NBLK_A, B_IMG), BLK, 0, stream>>>(pred_bbs, pred_cs, tar_bbs, tar_c,
                                                          anchors, grid_size, gt_ovl, gt_idx,
                                                          p_loc, p_nll, p_np);
    k_final     <<<1, 32, 0, stream>>>(p_loc, p_nll, p_np, out);
}